// SelfAttention_77163382440771
// MI455X (gfx1250) — compile-verified
//
#include <hip/hip_runtime.h>

typedef float v2f __attribute__((ext_vector_type(2)));
typedef float v4f __attribute__((ext_vector_type(4)));
typedef float v8f __attribute__((ext_vector_type(8)));

constexpr int NS = 4096;          // in_chan == seq dim
constexpr int NB = 16;            // batch
constexpr int BS = NB * NS;       // elements per (B,S) matrix
constexpr int KSPLIT = 2;
constexpr int KCH = NS / KSPLIT;  // 2048 K per partial

#if defined(__gfx1250__) &&                                         \
    __has_builtin(__builtin_amdgcn_global_load_async_to_lds_b128) && \
    __has_builtin(__builtin_amdgcn_s_wait_asynccnt)
#define USE_ASYNC_LDS 1
#else
#define USE_ASYNC_LDS 0
#endif

#if USE_ASYNC_LDS
// Toolchain-probed signature: param 0 is AS1 (global) pointer to a 16-byte
// int vector (printed as 'vector_size(4*sizeof(int)) int __device__ *').
typedef int v4i __attribute__((vector_size(16)));
typedef __attribute__((address_space(1))) v4i* g_v4i_ptr;
typedef __attribute__((address_space(3))) v4i* l_v4i_ptr;
__device__ __forceinline__ void async_copy16(const float4* g, float4* l) {
  // global_load_async_to_lds_b128 (tracked by ASYNCcnt)
  __builtin_amdgcn_global_load_async_to_lds_b128((g_v4i_ptr)g, (l_v4i_ptr)l,
                                                 0, 0);
}
#endif

// ---------------------------------------------------------------------------
// Kernel 1: qkv = x @ W^T (+ bias on K-half 0), fp32 WMMA 16x16x4.
// One wave computes a full 16(M=batch) x 16(N=channels) tile over KCH.
// Per 8-K block: one b128 A load + one b128 B load -> two WMMAs.
// K mapping per WMMA: internal K{0,1} = lane-half0's pair, K{2,3} = half1's
// pair; A and B use the identical mapping, and sum order over K is free.
// blockDim=256 (8 waves) -> 128 channels per WG; grid=(NS/128, KSPLIT, 3).
// ---------------------------------------------------------------------------
__global__ __launch_bounds__(256)
void qkv_gemm_wmma(const float* __restrict__ x,
                   const float* __restrict__ wq, const float* __restrict__ bq,
                   const float* __restrict__ wk, const float* __restrict__ bk,
                   const float* __restrict__ wv, const float* __restrict__ bv,
                   float* __restrict__ qkv) {
  const int wave = threadIdx.x >> 5;
  const int lane = threadIdx.x & 31;
  const int hh   = lane >> 4;    // wave half (K grouping)
  const int lr   = lane & 15;    // row-in-tile index
  const int nbase = blockIdx.x * 128 + wave * 16;
  const int z = blockIdx.z;                    // 0=q 1=k 2=v
  const int kbase = blockIdx.y * KCH;

  const float* w    = (z == 0) ? wq : (z == 1) ? wk : wv;
  const float* bias = (z == 0) ? bq : (z == 1) ? bk : bv;
  float* out = qkv + (size_t)(blockIdx.y * 3 + z) * BS;

  // A fragment (x): lane -> M=lr, loads K = kbase+k+4*hh .. +3 (b128)
  const float* aptr = x + lr * NS + kbase + 4 * hh;
  // B fragment (W^T): B[k][n] = W[n][k]; lane -> N=lr, same K window
  const float* bptr = w + (size_t)(nbase + lr) * NS + kbase + 4 * hh;

  v8f acc = {};
#pragma unroll 8
  for (int k = 0; k < KCH; k += 8) {
    v4f a = *(const v4f*)(aptr + k);
    v4f b = *(const v4f*)(bptr + k);
    acc = __builtin_amdgcn_wmma_f32_16x16x4_f32(false, a.lo, false, b.lo,
                                                (short)0, acc, false, false);
    acc = __builtin_amdgcn_wmma_f32_16x16x4_f32(false, a.hi, false, b.hi,
                                                (short)0, acc, false, false);
  }

  // C/D layout: VGPR v -> rows M=v (lanes 0-15) and M=v+8 (lanes 16-31), N=lr
  const float bb = (blockIdx.y == 0) ? bias[nbase + lr] : 0.0f;
#pragma unroll
  for (int v = 0; v < 8; ++v) {
    const int m = v + 8 * hh;
    out[m * NS + nbase + lr] = acc[v] + bb;
  }
}

// ---------------------------------------------------------------------------
// Kernel 2: fused rank-1 attention. scores[b,i,j] = q_i * k_j, softmax over j,
// weighted sum of v. Never materializes (B,S,S). k,v staged in LDS (async
// global->LDS on gfx1250); per-row max from m_i = max(q_i*kmax, q_i*kmin).
// grid=(NS/256, NB), block=256: one thread per output row i.
// ---------------------------------------------------------------------------
__global__ __launch_bounds__(256)
void attention_rank1(const float* __restrict__ qkv, float* __restrict__ out) {
  __shared__ float kk[NS];
  __shared__ float vv[NS];
#if USE_ASYNC_LDS
  __shared__ float kk1s[NS];
  __shared__ float vv1s[NS];
#endif
  __shared__ float smax[256];
  __shared__ float smin[256];

  const int b   = blockIdx.y;
  const int tid = threadIdx.x;

  const float* q0 = qkv + 0 * BS + b * NS;
  const float* k0 = qkv + 1 * BS + b * NS;
  const float* v0 = qkv + 2 * BS + b * NS;
  const float* q1 = qkv + 3 * BS + b * NS;
  const float* k1 = qkv + 4 * BS + b * NS;
  const float* v1 = qkv + 5 * BS + b * NS;

  const float4* k0v = (const float4*)k0;
  const float4* k1v = (const float4*)k1;
  const float4* v0v = (const float4*)v0;
  const float4* v1v = (const float4*)v1;
  float4* kk4 = (float4*)kk;
  float4* vv4 = (float4*)vv;

  float lmax = -3.0e38f, lmin = 3.0e38f;

#if USE_ASYNC_LDS
  {
    float4* kk1v = (float4*)kk1s;
    float4* vv1v = (float4*)vv1s;
#pragma unroll
    for (int t = tid; t < NS / 4; t += 256) {
      async_copy16(k0v + t, kk4 + t);
      async_copy16(k1v + t, kk1v + t);
      async_copy16(v0v + t, vv4 + t);
      async_copy16(v1v + t, vv1v + t);
    }
    __builtin_amdgcn_s_wait_asynccnt(0);
    __syncthreads();
    // Combine K-split partials in place; track min/max of combined k.
#pragma unroll
    for (int t = tid; t < NS / 4; t += 256) {
      float4 ka = kk4[t], kb = kk1v[t];
      float4 va = vv4[t], vb = vv1v[t];
      float4 kc = {ka.x + kb.x, ka.y + kb.y, ka.z + kb.z, ka.w + kb.w};
      float4 vc = {va.x + vb.x, va.y + vb.y, va.z + vb.z, va.w + vb.w};
      kk4[t] = kc;
      vv4[t] = vc;
      lmax = fmaxf(fmaxf(lmax, kc.x), fmaxf(kc.y, fmaxf(kc.z, kc.w)));
      lmin = fminf(fminf(lmin, kc.x), fminf(kc.y, fminf(kc.z, kc.w)));
    }
  }
#else
#pragma unroll
  for (int t = tid; t < NS / 4; t += 256) {
    float4 ka = k0v[t], kb = k1v[t];
    float4 va = v0v[t], vb = v1v[t];
    float4 kc = {ka.x + kb.x, ka.y + kb.y, ka.z + kb.z, ka.w + kb.w};
    float4 vc = {va.x + vb.x, va.y + vb.y, va.z + vb.z, va.w + vb.w};
    kk4[t] = kc;
    vv4[t] = vc;
    lmax = fmaxf(fmaxf(lmax, kc.x), fmaxf(kc.y, fmaxf(kc.z, kc.w)));
    lmin = fminf(fminf(lmin, kc.x), fminf(kc.y, fminf(kc.z, kc.w)));
  }
#endif

  smax[tid] = lmax;
  smin[tid] = lmin;
  __syncthreads();
#pragma unroll
  for (int s = 128; s > 0; s >>= 1) {
    if (tid < s) {
      smax[tid] = fmaxf(smax[tid], smax[tid + s]);
      smin[tid] = fminf(smin[tid], smin[tid + s]);
    }
    __syncthreads();
  }
  const float kmax = smax[0];
  const float kmin = smin[0];

  // One row per thread: fused streaming softmax + weighted sum (LDS broadcast).
  const int i = blockIdx.x * 256 + tid;
  const float qi = q0[i] + q1[i];
  constexpr float LOG2E = 1.4426950408889634f;
  const float a  = qi * LOG2E;
  const float mb = fmaxf(qi * kmax, qi * kmin) * LOG2E;  // row max, base-2

  float den = 0.0f, num = 0.0f;
#pragma unroll 4
  for (int j = 0; j < NS / 4; ++j) {
    float4 kj = kk4[j];
    float4 vj = vv4[j];
    float e0 = __builtin_amdgcn_exp2f(a * kj.x - mb);
    float e1 = __builtin_amdgcn_exp2f(a * kj.y - mb);
    float e2 = __builtin_amdgcn_exp2f(a * kj.z - mb);
    float e3 = __builtin_amdgcn_exp2f(a * kj.w - mb);
    den += (e0 + e1) + (e2 + e3);
    num += (e0 * vj.x + e1 * vj.y) + (e2 * vj.z + e3 * vj.w);
  }
  out[b * NS + i] = num / den;
}

// ---------------------------------------------------------------------------
extern "C" void kernel_launch(void* const* d_in, const int* in_sizes, int n_in,
                              void* d_out, int out_size, void* d_ws, size_t ws_size,
                              hipStream_t stream) {
  const float* x  = (const float*)d_in[0];
  const float* wq = (const float*)d_in[1];
  const float* bq = (const float*)d_in[2];
  const float* wk = (const float*)d_in[3];
  const float* bk = (const float*)d_in[4];
  const float* wv = (const float*)d_in[5];
  const float* bv = (const float*)d_in[6];
  float* qkv = (float*)d_ws;  // [KSPLIT][3][NB][NS] = 1.5 MiB fp32 partials

  qkv_gemm_wmma<<<dim3(NS / 128, KSPLIT, 3), 256, 0, stream>>>(
      x, wq, bq, wk, bk, wv, bv, qkv);
  attention_rank1<<<dim3(NS / 256, NB), 256, 0, stream>>>(qkv, (float*)d_out);
}